// GiddLoss_58377195487468
// MI455X (gfx1250) — compile-verified
//
#include <hip/hip_runtime.h>

// GIDD loss, MI455X (gfx1250, wave32). One workgroup per sequence position.
// TDM (tensor_load_to_lds) DMAs the full 32000-float logits row (128 KB) into
// LDS once -> single HBM pass (~262 MB total, ~11 us roofline at 23.3 TB/s).
// q-distribution side is closed-form (one-hot + uniform + mask spike), so the
// only O(V) work is max, sum(exp), and sum(log(a*e + c + eps)) — done with
// float4 LDS sweeps and raw v_exp_f32/v_log_f32 so compute stays under the
// HBM roofline.

#define VOCAB 32000
#define VOCAB4 (VOCAB / 4)
#define P_UNIFORM 0.05f
#define LOGIT_NEG (-1.0e6f)
#define GIDD_EPS 1e-12f

typedef __attribute__((ext_vector_type(4))) unsigned int v4u;
typedef __attribute__((ext_vector_type(8))) int v8i;
typedef __attribute__((ext_vector_type(4))) int v4i;

static constexpr int BLOCK = 512;
static constexpr int NWAVES = BLOCK / 32;

__device__ __forceinline__ float wave_sum32(float v) {
    #pragma unroll
    for (int o = 16; o > 0; o >>= 1) v += __shfl_xor(v, o, 32);
    return v;
}
__device__ __forceinline__ float wave_max32(float v) {
    #pragma unroll
    for (int o = 16; o > 0; o >>= 1) v = fmaxf(v, __shfl_xor(v, o, 32));
    return v;
}

__device__ __forceinline__ float block_sum(float v, float* red, int tid) {
    v = wave_sum32(v);
    if ((tid & 31) == 0) red[tid >> 5] = v;
    __syncthreads();
    if (tid < 32) {
        float x = (tid < NWAVES) ? red[tid] : 0.0f;
        x = wave_sum32(x);
        if (tid == 0) red[0] = x;
    }
    __syncthreads();
    float r = red[0];
    __syncthreads();
    return r;
}

__device__ __forceinline__ float block_max(float v, float* red, int tid) {
    v = wave_max32(v);
    if ((tid & 31) == 0) red[tid >> 5] = v;
    __syncthreads();
    if (tid < 32) {
        float x = (tid < NWAVES) ? red[tid] : -3.4e38f;
        x = wave_max32(x);
        if (tid == 0) red[0] = x;
    }
    __syncthreads();
    float r = red[0];
    __syncthreads();
    return r;
}

__global__ __launch_bounds__(BLOCK) void gidd_loss_kernel(
    const float* __restrict__ logits,
    const int* __restrict__ input_ids,
    const int* __restrict__ labels,
    const float* __restrict__ log_snr,
    float* __restrict__ out, int npos)
{
    __shared__ float row[VOCAB];   // 128000 B: full logits row, later exp(l-m)
    __shared__ float red[NWAVES];

    const int pos = blockIdx.x;
    if (pos >= npos) return;
    const int tid = threadIdx.x;
    float4* __restrict__ row4 = (float4*)row;

    // ---- TDM: async DMA of one 32000-float row into LDS (wave 0 only;
    // TDM ignores EXEC, so other waves must branch around it entirely). ----
    if (tid < 32) {
        unsigned long long gaddr =
            (unsigned long long)(const void*)(logits + (size_t)pos * VOCAB);
        // low 32 bits of the flat shared-aperture address == LDS byte offset
        unsigned int laddr =
            (unsigned int)(unsigned long long)(const void*)&row[0];

        v4u g0;
        g0[0] = 1u;                                   // count=1 (valid), user mode
        g0[1] = laddr;                                // lds_addr [63:32]
        g0[2] = (unsigned int)(gaddr & 0xffffffffu);  // global_addr [95:64]
        g0[3] = ((unsigned int)(gaddr >> 32) & 0x01ffffffu) // global_addr [120:96]
              | 0x80000000u;                          // type=2 ("image") [127:126]

        v8i g1;
        g1[0] = 0x00020000;                    // workgroup_mask=0, data_size=2 (4B)
        g1[1] = (int)((VOCAB & 0xffff) << 16); // tensor_dim0[15:0] -> bits[63:48]
        g1[2] = (int)(((VOCAB >> 16) & 0xffff) | (1u << 16)); // dim0 hi | tensor_dim1=1
        g1[3] = (int)((VOCAB & 0xffff) << 16); // tile_dim0=32000 -> bits[127:112]
        g1[4] = 1;                             // tile_dim1=1, tile_dim2=0
        g1[5] = VOCAB;                         // tensor_dim0_stride low 32
        g1[6] = 0;                             // stride hi, tensor_dim1_stride lo
        g1[7] = 0;

        v4i gz4 = {0, 0, 0, 0};                  // groups 2/3 unused (<=2D tensor)
        v8i gz8 = {0, 0, 0, 0, 0, 0, 0, 0};      // extra group (clang-23 form), unused
        __builtin_amdgcn_tensor_load_to_lds(g0, g1, gz4, gz4, gz8, 0);
        __builtin_amdgcn_s_wait_tensorcnt(0);
    }
    __syncthreads();

    // Reference sets logits[..., -1] = -1e6 before softmax.
    if (tid == 0) row[VOCAB - 1] = LOGIT_NEG;
    __syncthreads();

    // ---- pass 1: row max (float4 LDS sweep) ----
    float m = -3.4e38f;
    for (int v = tid; v < VOCAB4; v += BLOCK) {
        float4 l = row4[v];
        m = fmaxf(m, fmaxf(fmaxf(l.x, l.y), fmaxf(l.z, l.w)));
    }
    m = block_max(m, red, tid);

    // ---- pass 2: e = exp(l - m); Z = sum(e); store e back into LDS ----
    float zs = 0.0f;
    for (int v = tid; v < VOCAB4; v += BLOCK) {
        float4 l = row4[v];
        float4 e;
        e.x = __expf(l.x - m);
        e.y = __expf(l.y - m);
        e.z = __expf(l.z - m);
        e.w = __expf(l.w - m);
        row4[v] = e;
        zs += (e.x + e.y) + (e.z + e.w);
    }
    const float Z = block_sum(zs, red, tid);

    const float lsnr = log_snr[pos];
    const float alpha = 1.0f / (1.0f + __expf(-lsnr));
    const float oma   = 1.0f - alpha;
    const float c     = oma * (P_UNIFORM / (float)VOCAB); // uniform floor of p,q
    const float cpe   = c + GIDD_EPS;
    const float maskadd = oma * (1.0f - P_UNIFORM);       // mask spike
    const float a     = alpha / Z;                        // p_generic = a*e + c

    // ---- pass 3: S_gen = sum_v log(a*e_v + c + eps) ----
    float ls = 0.0f;
    for (int v = tid; v < VOCAB4; v += BLOCK) {
        float4 e = row4[v];
        float lx = __logf(fmaf(a, e.x, cpe));
        float ly = __logf(fmaf(a, e.y, cpe));
        float lz = __logf(fmaf(a, e.z, cpe));
        float lw = __logf(fmaf(a, e.w, cpe));
        ls += (lx + ly) + (lz + lw);
    }
    const float S_gen = block_sum(ls, red, tid);

    // ---- closed-form q side + gathers + final loss (thread 0) ----
    if (tid == 0) {
        const int MASK = VOCAB - 1;
        const int lab = labels[pos];
        const int z   = input_ids[pos];
        const float e_lab  = row[lab];
        const float e_z    = row[z];
        const float e_mask = row[MASK]; // == exp(-1e6 - m) == 0

        const float logp_gen_lab  = logf(fmaf(a, e_lab,  cpe));
        const float logp_gen_mask = logf(fmaf(a, e_mask, cpe));
        const float logp_mask     = logf(fmaf(a, e_mask, c) + maskadd + GIDD_EPS);

        float T_q, Wsum;
        if (lab != MASK) {
            const float q_lab  = alpha + c;
            const float q_mask = c + maskadd;
            Wsum = cpe * (S_gen - logp_gen_lab - logp_gen_mask)
                 + (q_lab  + GIDD_EPS) * logp_gen_lab
                 + (q_mask + GIDD_EPS) * logp_mask;
            T_q  = (float)(VOCAB - 2) * cpe * logf(cpe)
                 + (q_lab  + GIDD_EPS) * logf(q_lab  + GIDD_EPS)
                 + (q_mask + GIDD_EPS) * logf(q_mask + GIDD_EPS);
        } else {
            const float q_mask = alpha + c + maskadd;
            Wsum = cpe * (S_gen - logp_gen_mask)
                 + (q_mask + GIDD_EPS) * logp_mask;
            T_q  = (float)(VOCAB - 1) * cpe * logf(cpe)
                 + (q_mask + GIDD_EPS) * logf(q_mask + GIDD_EPS);
        }
        const float kl = T_q - Wsum;

        float q_z;
        if (z == MASK) q_z = (lab == MASK) ? (alpha + c + maskadd) : (c + maskadd);
        else           q_z = (z == lab)    ? (alpha + c)           : c;
        const float p_z = fmaf(a, e_z, c) + ((z == MASK) ? maskadd : 0.0f);

        const float qv = q_z + GIDD_EPS;
        const float pv = p_z + GIDD_EPS;
        const float is_div = qv / (pv + GIDD_EPS) - (logf(qv) - logf(pv)) - 1.0f;

        const float w = alpha * (1.0f - alpha);
        const float loss_w = fminf(fmaxf(w, 0.0f), 1.0e3f);
        out[pos] = loss_w * (kl + is_div);
    }
}

extern "C" void kernel_launch(void* const* d_in, const int* in_sizes, int n_in,
                              void* d_out, int out_size, void* d_ws, size_t ws_size,
                              hipStream_t stream) {
    const float* logits    = (const float*)d_in[0];
    const int*   input_ids = (const int*)d_in[1];
    const int*   labels    = (const int*)d_in[2];
    const float* log_snr   = (const float*)d_in[3];
    float* out = (float*)d_out;
    const int npos = in_sizes[3]; // B*S
    gidd_loss_kernel<<<dim3(npos), dim3(BLOCK), 0, stream>>>(
        logits, input_ids, labels, log_snr, out, npos);
}